// GraphSAGEModel_32976758899207
// MI455X (gfx1250) — compile-verified
//
#include <hip/hip_runtime.h>

#define N_NODES 100000
#define N_EDGES 1600000
#define D 128
#define BN_EPS 1e-5f

typedef __attribute__((ext_vector_type(2))) float v2f;
typedef __attribute__((ext_vector_type(8))) float v8f;

// ---------------------------------------------------------------- utilities
__global__ void zero_f32(float* __restrict__ p, long n) {
    long i = (long)blockIdx.x * blockDim.x + threadIdx.x;
    if (i < n) p[i] = 0.0f;
}

// ---------------------------------------------------------------- degree
__global__ void degree_kernel(const int* __restrict__ ei, float* __restrict__ deg) {
    int e = blockIdx.x * blockDim.x + threadIdx.x;
    if (e < N_EDGES) {
        int dst = ei[N_EDGES + e];
        atomicAdd(&deg[dst], 1.0f);
    }
}

__global__ void deginv_kernel(float* __restrict__ deg) {
    int i = blockIdx.x * blockDim.x + threadIdx.x;
    if (i < N_NODES) deg[i] = 1.0f / fmaxf(deg[i], 1.0f);
}

// ---------------------------------------------------------------- scatter-add
// 32 lanes per edge, 4 channels (float4) per lane: coalesced 512B gather of the
// source row, then 4 global f32 atomic adds into the destination row.
__global__ void scatter_kernel(const float* __restrict__ x,
                               const int* __restrict__ ei,
                               float* __restrict__ agg) {
    long tid = (long)blockIdx.x * blockDim.x + threadIdx.x;
    long e = tid >> 5;
    if (e >= N_EDGES) return;
    int c = ((int)tid & 31) * 4;
    int src = ei[e];
    int dst = ei[N_EDGES + e];
    const float4 v = *(const float4*)(x + (long)src * D + c);
    float* out = agg + (long)dst * D + c;
    atomicAdd(out + 0, v.x);
    atomicAdd(out + 1, v.y);
    atomicAdd(out + 2, v.z);
    atomicAdd(out + 3, v.w);
}

// ---------------------------------------------------------------- SAGE GEMM
// h[m, n] = sum_k aggS[m,k]*Wl[k,n] + sum_k x[m,k]*Wr[k,n] + bl[n]
// Persistent blocks: 625 blocks x 10 M-tiles. Each block stages Wl/Wr ONCE in
// LDS (transposed, sW[n][k]) so both A and B WMMA fragments are contiguous
// (ka, ka+1) pairs -> ds_load_2addr_b64, zero global loads in the K loop.
// Row stride 132 (bank = 4*row + k mod 64) -> conflict-free fragment reads.
#define SROW 132
#define NBLK 625
#define TILES_PER_BLK 10  // 625 * 10 * 16 rows = 100000 nodes

template <bool WITH_STATS>
__global__ __launch_bounds__(256) void sage_gemm(
    const float* __restrict__ x, const float* __restrict__ agg,
    const float* __restrict__ deg_inv,
    const float* __restrict__ Wl, const float* __restrict__ Wr,
    const float* __restrict__ bl,
    float* __restrict__ h, float* __restrict__ stats) {
    __shared__ float sWl[D * SROW];   // transposed: sWl[n*SROW + k] = Wl[k, n]
    __shared__ float sWr[D * SROW];
    __shared__ float sA[16 * SROW];   // (agg * deg_inv) slab, row-major
    __shared__ float sX[16 * SROW];   // x slab, row-major

    // ---- stage weights once per block (coalesced float4 reads, transpose) ----
    for (int i = threadIdx.x; i < (D * D) / 4; i += 256) {
        int idx = i * 4;
        int k = idx >> 7;          // weight row
        int n = idx & (D - 1);     // weight col (start of 4)
        float4 wl4 = *(const float4*)(Wl + idx);
        float4 wr4 = *(const float4*)(Wr + idx);
        sWl[(n + 0) * SROW + k] = wl4.x;
        sWl[(n + 1) * SROW + k] = wl4.y;
        sWl[(n + 2) * SROW + k] = wl4.z;
        sWl[(n + 3) * SROW + k] = wl4.w;
        sWr[(n + 0) * SROW + k] = wr4.x;
        sWr[(n + 1) * SROW + k] = wr4.y;
        sWr[(n + 2) * SROW + k] = wr4.z;
        sWr[(n + 3) * SROW + k] = wr4.w;
    }

    const int lane = threadIdx.x & 31;
    const int wave = threadIdx.x >> 5;
    const int n0 = wave * 16;
    const int half = lane >> 4;   // 0: K pair {k0,k0+1}; 1: K pair {k0+2,k0+3}
    const int l15 = lane & 15;    // A row / B,C column within tile
    const int wrow = (n0 + l15) * SROW;   // this lane's column slab in sW*
    const float bias = bl[n0 + l15];

    for (int tt = 0; tt < TILES_PER_BLK; ++tt) {
        const int m0 = (blockIdx.x * TILES_PER_BLK + tt) * 16;

        // barrier also protects sA/sX from readers of the previous tile
        __syncthreads();

        // ---- stage 16x128 A (scaled agg) and X slabs (float4, coalesced) ----
        for (int i = threadIdx.x; i < (16 * D) / 4; i += 256) {
            int idx = i * 4;
            int r = idx >> 7;
            int c = idx & (D - 1);
            long g = (long)(m0 + r) * D + c;
            float di = deg_inv[m0 + r];
            float4 a4 = *(const float4*)(agg + g);
            float4 x4 = *(const float4*)(x + g);
            *(float4*)(sA + r * SROW + c) =
                make_float4(a4.x * di, a4.y * di, a4.z * di, a4.w * di);
            *(float4*)(sX + r * SROW + c) = x4;
        }
        __syncthreads();

        // ---- K loop: all operands from LDS ----
        v8f acc = {};
        for (int k0 = 0; k0 < D; k0 += 4) {
            const int ka = k0 + 2 * half;
            v2f a, b;
            a.x = sA[l15 * SROW + ka];
            a.y = sA[l15 * SROW + ka + 1];
            b.x = sWl[wrow + ka];
            b.y = sWl[wrow + ka + 1];
            acc = __builtin_amdgcn_wmma_f32_16x16x4_f32(false, a, false, b,
                                                        (short)0, acc, false, false);
            a.x = sX[l15 * SROW + ka];
            a.y = sX[l15 * SROW + ka + 1];
            b.x = sWr[wrow + ka];
            b.y = sWr[wrow + ka + 1];
            acc = __builtin_amdgcn_wmma_f32_16x16x4_f32(false, a, false, b,
                                                        (short)0, acc, false, false);
        }

        // ---- epilogue: bias, store, fused per-channel BN partial sums ----
        float s = 0.0f, s2 = 0.0f;
#pragma unroll
        for (int r = 0; r < 8; ++r) {
            float v = acc[r] + bias;
            int row = m0 + r + 8 * half;       // C/D layout: VGPR r -> M = r (+8)
            h[(long)row * D + n0 + l15] = v;
            if (WITH_STATS) { s += v; s2 += v * v; }
        }
        if (WITH_STATS) {
            // lanes l and l+16 hold the same column -> fold halves, 1 atomic each
            s  += __shfl_xor(s, 16, 32);
            s2 += __shfl_xor(s2, 16, 32);
            if (half == 0) {
                atomicAdd(&stats[n0 + l15], s);
                atomicAdd(&stats[D + n0 + l15], s2);
            }
        }
    }
}

// ---------------------------------------------------------------- BN + ReLU
__global__ void bn_relu_kernel(float* __restrict__ h,
                               const float* __restrict__ stats,
                               const float* __restrict__ gamma,
                               const float* __restrict__ beta) {
    long i = (long)blockIdx.x * blockDim.x + threadIdx.x;
    if (i >= (long)N_NODES * D) return;
    int c = (int)(i & (D - 1));
    const float invN = 1.0f / (float)N_NODES;
    float mu  = stats[c] * invN;
    float var = stats[D + c] * invN - mu * mu;
    float sc  = rsqrtf(var + BN_EPS) * gamma[c];
    float sh  = beta[c] - mu * sc;
    float v = h[i] * sc + sh;
    h[i] = v > 0.0f ? v : 0.0f;
}

// ---------------------------------------------------------------- launcher
extern "C" void kernel_launch(void* const* d_in, const int* in_sizes, int n_in,
                              void* d_out, int out_size, void* d_ws, size_t ws_size,
                              hipStream_t stream) {
    (void)in_sizes; (void)n_in; (void)out_size; (void)ws_size;
    const float* x     = (const float*)d_in[0];
    const int*   ei    = (const int*)d_in[1];   // [2, N_EDGES]
    const float* Wl    = (const float*)d_in[2]; // [3, D, D]
    const float* bl    = (const float*)d_in[3]; // [3, D]
    const float* Wr    = (const float*)d_in[4]; // [3, D, D]
    const float* gamma = (const float*)d_in[5]; // [2, D]
    const float* beta  = (const float*)d_in[6]; // [2, D]
    float* out = (float*)d_out;

    // Workspace layout (floats): deg_inv | stats | agg | bufA | bufB
    float* ws    = (float*)d_ws;
    float* deg   = ws;                                   // N_NODES
    float* stats = ws + N_NODES;                         // 2*D
    float* agg   = stats + 2 * D;                        // N_NODES*D
    float* bufA  = agg + (long)N_NODES * D;              // N_NODES*D
    float* bufB  = bufA + (long)N_NODES * D;             // N_NODES*D

    const long ND = (long)N_NODES * D;
    const int TPB = 256;

    // Degree (shared by all layers)
    zero_f32<<<(N_NODES + TPB - 1) / TPB, TPB, 0, stream>>>(deg, N_NODES);
    degree_kernel<<<(N_EDGES + TPB - 1) / TPB, TPB, 0, stream>>>(ei, deg);
    deginv_kernel<<<(N_NODES + TPB - 1) / TPB, TPB, 0, stream>>>(deg);

    const float* xin[3]  = {x, bufA, bufB};
    float*       hout[3] = {bufA, bufB, out};

    for (int l = 0; l < 3; ++l) {
        zero_f32<<<(int)((ND + TPB - 1) / TPB), TPB, 0, stream>>>(agg, ND);
        zero_f32<<<1, TPB, 0, stream>>>(stats, 2 * D);
        scatter_kernel<<<(int)(((long)N_EDGES * 32 + TPB - 1) / TPB), TPB, 0, stream>>>(
            xin[l], ei, agg);
        if (l < 2) {
            sage_gemm<true><<<NBLK, TPB, 0, stream>>>(
                xin[l], agg, deg, Wl + l * D * D, Wr + l * D * D, bl + l * D,
                hout[l], stats);
            bn_relu_kernel<<<(int)((ND + TPB - 1) / TPB), TPB, 0, stream>>>(
                hout[l], stats, gamma + l * D, beta + l * D);
        } else {
            sage_gemm<false><<<NBLK, TPB, 0, stream>>>(
                xin[l], agg, deg, Wl + l * D * D, Wr + l * D * D, bl + l * D,
                hout[l], stats);
        }
    }
}